// UnFusedGCN_21543555956849
// MI455X (gfx1250) — compile-verified
//
#include <hip/hip_runtime.h>

typedef float v2f __attribute__((ext_vector_type(2)));
typedef float v8f __attribute__((ext_vector_type(8)));

#define GCN_N_NODES 50000
#define GCN_FEAT    128
#define GCN_EMBED   128
#define GCN_NCLS    64
#define GCN_N_EDGES 800000

// ---------------------------------------------------------------------------
// Dense GEMM + bias (+ optional fused ReLU on the A operand) using the
// full-precision CDNA5 matrix op V_WMMA_F32_16X16X4_F32.
//   H[N, OUT] = act(A[N, 128]) @ W[128, OUT] + bias
// One block = one 16-row tile of A staged in LDS; wave y owns a 16-col tile.
// N_NODES = 50000 = 3125 * 16 exactly -> no row remainder, EXEC all-ones.
// ---------------------------------------------------------------------------
template <int OUT, bool RELU>
__global__ __launch_bounds__(32 * (OUT / 16))
void gcn_gemm_bias_wmma(const float* __restrict__ A,    // [N, 128]
                        const float* __restrict__ W,    // [128, OUT]
                        const float* __restrict__ bias, // [OUT]
                        float* __restrict__ H)          // [N, OUT]
{
    constexpr int K      = GCN_FEAT;          // 128
    constexpr int NTHR   = 32 * (OUT / 16);
    constexpr int LDSTRD = 132;               // 128 + 4 pad: conflict-free A reads

    __shared__ float As[16 * LDSTRD];

    const int lane    = threadIdx.x;          // 0..31 (wave32)
    const int waveCol = threadIdx.y;          // which 16-col tile
    const int rowBase = blockIdx.x * 16;
    const int tid     = waveCol * 32 + lane;

    // Cooperative stage of the 16x128 A tile (ReLU fused here for layer 2).
    for (int i = tid; i < 16 * K; i += NTHR) {
        const int r = i >> 7;                 // /128
        const int c = i & 127;
        float v = A[(size_t)(rowBase + r) * K + c];
        if (RELU) v = fmaxf(v, 0.0f);
        As[r * LDSTRD + c] = v;
    }
    __syncthreads();

    const int half = lane >> 4;               // 0: lanes 0-15, 1: lanes 16-31
    const int m    = lane & 15;               // A row within tile / B,N column
    const int col  = waveCol * 16 + m;        // output column

    v8f acc = {};
    #pragma unroll
    for (int k = 0; k < K; k += 4) {
        const int ka = k + 2 * half;          // lane-half selects K {0,1} vs {2,3}
        v2f a, b;
        // A fragment: 16x4 f32 (ISA: lane=M, VGPR0=Ka, VGPR1=Ka+1)
        a.x = As[m * LDSTRD + ka];
        a.y = As[m * LDSTRD + ka + 1];
        // B fragment: 4x16 f32 (rows striped across lanes, same half-split)
        b.x = W[(size_t)(ka)     * OUT + col];
        b.y = W[(size_t)(ka + 1) * OUT + col];
        acc = __builtin_amdgcn_wmma_f32_16x16x4_f32(
            /*neg_a=*/false, a, /*neg_b=*/false, b,
            /*c_mod=*/(short)0, acc, /*reuse_a=*/false, /*reuse_b=*/false);
    }

    const float bv = bias[col];
    #pragma unroll
    for (int r = 0; r < 8; ++r) {
        const int row = r + half * 8;         // C/D layout: VGPR r -> M = r or r+8
        H[(size_t)(rowBase + row) * OUT + col] = acc[r] + bv;
    }
}

// ---------------------------------------------------------------------------
// Edge scatter: out[dst] += val * H[src], OUT/4 lanes per edge, float4 gather
// (L2-resident since H fits in the 192 MB L2), f32 global atomics for scatter.
// ---------------------------------------------------------------------------
template <int OUT>
__global__ __launch_bounds__(256)
void gcn_spmm_scatter(const float* __restrict__ H,
                      const int*   __restrict__ src,
                      const int*   __restrict__ dst,
                      const float* __restrict__ val,
                      float* __restrict__ out,
                      int nEdges)
{
    constexpr int TPE = OUT / 4;              // threads per edge (32 or 16)
    const int tid = blockIdx.x * blockDim.x + threadIdx.x;
    const int e   = tid / TPE;
    const int f   = (tid % TPE) * 4;
    if (e >= nEdges) return;

    const int   s = src[e];
    const int   d = dst[e];
    const float v = val[e];

    const float4 hv = *(const float4*)(H + (size_t)s * OUT + f);
    float* op = out + (size_t)d * OUT + f;
    atomicAdd(op + 0, hv.x * v);
    atomicAdd(op + 1, hv.y * v);
    atomicAdd(op + 2, hv.z * v);
    atomicAdd(op + 3, hv.w * v);
}

// ---------------------------------------------------------------------------
// Pipeline:
//   zero(AGG1); H1 = x@W1+b1 (WMMA); AGG1 += scatter(H1);
//   G2 = relu(AGG1)@W2+b2 (WMMA, ReLU fused, G2 aliases H1's scratch);
//   zero(out); out += scatter(G2)
// Workspace: 2 * 50000 * 128 * 4 B = 51.2 MB.
// ---------------------------------------------------------------------------
extern "C" void kernel_launch(void* const* d_in, const int* in_sizes, int n_in,
                              void* d_out, int out_size, void* d_ws, size_t ws_size,
                              hipStream_t stream)
{
    const float* x    = (const float*)d_in[0];
    const int*   esrc = (const int*)  d_in[1];
    const int*   edst = (const int*)  d_in[2];
    const float* eval_= (const float*)d_in[3];
    const float* W1   = (const float*)d_in[4];
    const float* b1   = (const float*)d_in[5];
    const float* W2   = (const float*)d_in[6];
    const float* b2   = (const float*)d_in[7];
    float*       out  = (float*)d_out;

    float* H1   = (float*)d_ws;                        // 50000*128
    float* AGG1 = H1 + (size_t)GCN_N_NODES * GCN_EMBED; // 50000*128
    float* G2   = H1;                                   // reuse: H1 dead after spmm1

    const int rowTiles = GCN_N_NODES / 16;              // 3125 exactly

    // Layer 1
    hipMemsetAsync(AGG1, 0, (size_t)GCN_N_NODES * GCN_EMBED * sizeof(float), stream);
    gcn_gemm_bias_wmma<GCN_EMBED, false>
        <<<rowTiles, dim3(32, GCN_EMBED / 16), 0, stream>>>(x, W1, b1, H1);
    {
        const int total  = GCN_N_EDGES * (GCN_EMBED / 4);
        gcn_spmm_scatter<GCN_EMBED>
            <<<(total + 255) / 256, 256, 0, stream>>>(H1, esrc, edst, eval_, AGG1, GCN_N_EDGES);
    }

    // Layer 2 (ReLU fused into A staging)
    gcn_gemm_bias_wmma<GCN_NCLS, true>
        <<<rowTiles, dim3(32, GCN_NCLS / 16), 0, stream>>>(AGG1, W2, b2, G2);
    hipMemsetAsync(out, 0, (size_t)GCN_N_NODES * GCN_NCLS * sizeof(float), stream);
    {
        const int total  = GCN_N_EDGES * (GCN_NCLS / 4);
        gcn_spmm_scatter<GCN_NCLS>
            <<<(total + 255) / 256, 256, 0, stream>>>(G2, esrc, edst, eval_, out, GCN_N_EDGES);
    }
}